// StateSpaceLayer_19894288515300
// MI455X (gfx1250) — compile-verified
//
#include <hip/hip_runtime.h>
#include <math.h>

// Problem constants (from reference setup_inputs)
#define BDIM 8
#define FDIM 256
#define HDIM 64
#define TDIM 512
#define SDIM 32

typedef float v2f __attribute__((ext_vector_type(2)));
typedef float v4f __attribute__((ext_vector_type(4)));
typedef float v8f __attribute__((ext_vector_type(8)));

// ---------------------------------------------------------------------------
// Kernel 1: state trajectory  s_t = A @ s_{t-1} + Bvec, s_0 = 0  -> states[T,S]
// One wave32; lane i owns state element i; A row i cached in registers.
// ---------------------------------------------------------------------------
__global__ void __launch_bounds__(SDIM)
state_traj_kernel(const float* __restrict__ A, const float* __restrict__ Bvec,
                  float* __restrict__ states) {
    const int i = threadIdx.x;          // 0..31
    __shared__ float s[SDIM];
    float arow[SDIM];
#pragma unroll
    for (int j = 0; j < SDIM; ++j) arow[j] = A[i * SDIM + j];
    const float bi = Bvec[i];
    s[i] = 0.0f;
    __syncthreads();
    for (int t = 0; t < TDIM; ++t) {
        float acc = bi;
#pragma unroll
        for (int j = 0; j < SDIM; ++j) acc += arow[j] * s[j];
        __syncthreads();                // all lanes done reading old state
        s[i] = acc;
        states[t * SDIM + i] = acc;
        __syncthreads();                // new state visible before next step
    }
}

// ---------------------------------------------------------------------------
// Kernel 2: yc_tr[F,T] = Cmat^T (FxS) @ states^T (SxT)   (fp32 WMMA 16x16x4)
// Transposed so kernel 3 can read yc along t with float4 loads.
// One wave per 16x16 output tile (M = f, N = t); K = 32 -> 8 WMMA steps.
// A-frag (16x4):  lanes 0-15 hold {K0,K1} of row m, lanes 16-31 hold {K2,K3}.
// B-frag (4x16):  VGPR0 = rows K0|K2 across lanes, VGPR1 = rows K1|K3.
// D (16x16 f32):  VGPR r -> row r (lanes 0-15) / row r+8 (lanes 16-31).
// ---------------------------------------------------------------------------
__global__ void __launch_bounds__(32)
yc_gemm_kernel(const float* __restrict__ states, const float* __restrict__ Cmat,
               float* __restrict__ yc_tr) {
    const int lane = threadIdx.x;
    const int half = lane >> 4;         // 0: lanes 0-15, 1: lanes 16-31
    const int l15  = lane & 15;
    const int f0 = blockIdx.x * 16;     // gridDim.x = F/16 = 16
    const int t0 = blockIdx.y * 16;     // gridDim.y = T/16 = 32

    v8f acc = {};
#pragma unroll
    for (int k0 = 0; k0 < SDIM; k0 += 4) {
        const int koff = k0 + half * 2;
        v2f a, b;
        // A[m][k] = Cmat[k][f0+m]
        a.x = Cmat[(koff)     * FDIM + f0 + l15];
        a.y = Cmat[(koff + 1) * FDIM + f0 + l15];
        // B[k][n] = states[t0+n][k]
        b.x = states[(t0 + l15) * SDIM + koff];
        b.y = states[(t0 + l15) * SDIM + koff + 1];
        acc = __builtin_amdgcn_wmma_f32_16x16x4_f32(
            /*neg_a=*/false, a, /*neg_b=*/false, b,
            /*c_mod=*/(short)0, acc, /*reuse_a=*/false, /*reuse_b=*/false);
    }
#pragma unroll
    for (int r = 0; r < 8; ++r) {
        const int f = f0 + r + half * 8;
        const int t = t0 + l15;
        yc_tr[f * TDIM + t] = acc[r];   // coalesced over t
    }
}

// ---------------------------------------------------------------------------
// Kernel 3: fused  out = LN_F( x + gelu(yc[t,:] + x * D) )
// Block = one (b,h) x 32 consecutive t. 256 threads = 8 waves.
// Lane = (fr 0..3, tq 0..7): each lane owns 4 consecutive t as float4.
// Per iteration a wave covers 4 channels x 32 t -> b128 loads/stores
// (4 coalesced 128B segments per instruction). Intermediate kept in LDS
// (b128, conflict-free) so HBM traffic stays at read-x + write-out.
// x / out are streamed with non-temporal hints; yc_tr (512 KB, reused
// 512x) stays hot in the 192 MB L2.
// ---------------------------------------------------------------------------
__global__ void __launch_bounds__(256)
fuse_gelu_ln_kernel(const float* __restrict__ x, const float* __restrict__ yc_tr,
                    const float* __restrict__ Dv, const float* __restrict__ lnw,
                    const float* __restrict__ lnb, float* __restrict__ out) {
    const int TPOS = 32;
    const int tch  = TDIM / TPOS;               // 16
    const int blk  = blockIdx.x;                // B*H*tch = 8192
    const int t0   = (blk % tch) * TPOS;
    const int bh   = blk / tch;
    const int h    = bh % HDIM;
    const int b    = bh / HDIM;

    const int tid  = threadIdx.x;
    const int w    = tid >> 5;                  // wave 0..7
    const int lane = tid & 31;
    const int fr   = lane >> 3;                 // channel sub-row 0..3
    const int tq   = lane & 7;                  // t quad 0..7
    const int tl4  = tq * 4;                    // t offset within tile

    __shared__ __align__(16) float vals[FDIM * TPOS];  // 32 KB, [f*32 + t]
    __shared__ __align__(16) float psum[32][TPOS];     // 4 KB
    __shared__ __align__(16) float psq[32][TPOS];      // 4 KB
    __shared__ __align__(16) float mu_s[TPOS];
    __shared__ __align__(16) float rs_s[TPOS];

    const int fstride = HDIM * TDIM;            // 32768
    const int base = ((b * FDIM) * HDIM + h) * TDIM + t0 + tl4;  // f=0 quad

    v4f sum = {0.0f, 0.0f, 0.0f, 0.0f};
    v4f sq  = {0.0f, 0.0f, 0.0f, 0.0f};
#pragma unroll
    for (int k = 0; k < 8; ++k) {
        const int f = w * 32 + k * 4 + fr;
        const v4f xv = __builtin_nontemporal_load((const v4f*)(x + base + f * fstride));
        const v4f yv = *(const v4f*)(yc_tr + f * TDIM + t0 + tl4);
        const float d = Dv[f];
        v4f o;
#pragma unroll
        for (int j = 0; j < 4; ++j) {
            const float v = yv[j] + xv[j] * d;
            // exact GELU: 0.5*v*(1+erf(v/sqrt(2)))
            const float g = 0.5f * v * (1.0f + erff(v * 0.70710678118654752f));
            o[j] = xv[j] + g;
            sum[j] += o[j];
            sq[j]  += o[j] * o[j];
        }
        *(v4f*)(vals + f * TPOS + tl4) = o;     // ds_store_b128
    }
    const int c = w * 4 + fr;                   // contributor id 0..31
    *(v4f*)(&psum[c][tl4]) = sum;
    *(v4f*)(&psq[c][tl4])  = sq;
    __syncthreads();

    if (tid < 32) {                             // one t per lane
        float s = 0.0f, q = 0.0f;
#pragma unroll
        for (int cc = 0; cc < 32; ++cc) { s += psum[cc][tid]; q += psq[cc][tid]; }
        const float mu  = s * (1.0f / (float)FDIM);
        const float var = q * (1.0f / (float)FDIM) - mu * mu;
        mu_s[tid] = mu;
        rs_s[tid] = rsqrtf(var + 1e-5f);
    }
    __syncthreads();

    const v4f mu = *(const v4f*)(mu_s + tl4);
    const v4f rs = *(const v4f*)(rs_s + tl4);
#pragma unroll
    for (int k = 0; k < 8; ++k) {
        const int f = w * 32 + k * 4 + fr;
        const v4f val = *(const v4f*)(vals + f * TPOS + tl4);   // ds_load_b128
        const float wgt = lnw[f];
        const float bia = lnb[f];
        v4f o;
#pragma unroll
        for (int j = 0; j < 4; ++j)
            o[j] = (val[j] - mu[j]) * rs[j] * wgt + bia;
        __builtin_nontemporal_store(o, (v4f*)(out + base + f * fstride));
    }
}

// ---------------------------------------------------------------------------
extern "C" void kernel_launch(void* const* d_in, const int* in_sizes, int n_in,
                              void* d_out, int out_size, void* d_ws, size_t ws_size,
                              hipStream_t stream) {
    const float* x    = (const float*)d_in[0];   // [B,F,H,T]
    const float* A    = (const float*)d_in[1];   // [S,S]
    const float* Bvec = (const float*)d_in[2];   // [S]
    const float* Cmat = (const float*)d_in[3];   // [S,F]
    const float* Dv   = (const float*)d_in[4];   // [F]
    const float* lnw  = (const float*)d_in[5];   // [F]
    const float* lnb  = (const float*)d_in[6];   // [F]
    float* out = (float*)d_out;

    float* states = (float*)d_ws;                // [T,S]   = 64 KB
    float* yc_tr  = states + TDIM * SDIM;        // [F,T]   = 512 KB

    state_traj_kernel<<<1, SDIM, 0, stream>>>(A, Bvec, states);

    dim3 g2(FDIM / 16, TDIM / 16);               // 16 x 32 tiles (M=f, N=t)
    yc_gemm_kernel<<<g2, 32, 0, stream>>>(states, Cmat, yc_tr);

    const int nblk = BDIM * HDIM * (TDIM / 32);  // 8192
    fuse_gelu_ln_kernel<<<nblk, 256, 0, stream>>>(x, yc_tr, Dv, lnw, lnb, out);
}